// UHGGraphSAGELayer_30794915512954
// MI455X (gfx1250) — compile-verified
//
#include <hip/hip_runtime.h>
#include <hip/hip_bf16.h>

#define EPSF   1e-9f
#define WMINF  1e-6f
#define DIM    128
#define DP1    129

typedef __attribute__((ext_vector_type(2))) float v2f;
typedef __attribute__((ext_vector_type(8))) float v8f;

// ---------------------------------------------------------------------------
// Kernel 1: zero the accumulator (d_out doubles as neigh_sum[.,0:128] +
// neigh_weights[.,128]); harness poisons buffers, so init every call.
// ---------------------------------------------------------------------------
__global__ void uhg_zero_f32(float* __restrict__ p, long n) {
    long i = (long)blockIdx.x * blockDim.x + threadIdx.x;
    if (i < n) p[i] = 0.0f;
}

// ---------------------------------------------------------------------------
// Kernel 2: one wave32 per edge. Lanes cover dims [4*lane, 4*lane+3] of both
// endpoint rows; butterfly-reduce dot(129), |r|^2(128), |c|^2(128) in one
// pass; scatter w * features[col] with f32 global atomics. x (25.8 MB) is
// L2-resident on MI455X (192 MB L2), so gathers/atomics run at L2 speed.
// ---------------------------------------------------------------------------
__global__ __launch_bounds__(256) void uhg_edge_scatter(
    const float* __restrict__ x, const int* __restrict__ ei,
    float* __restrict__ acc, int E) {
    const int e = blockIdx.x * 8 + (threadIdx.x >> 5);
    if (e >= E) return;                       // uniform per wave
    const int lane = threadIdx.x & 31;
    const int row = ei[e];
    const int col = ei[E + e];
    const float* xr = x + (long)row * DP1;
    const float* xc = x + (long)col * DP1;
    const int d0 = 4 * lane;
    const float r0 = xr[d0 + 0], r1 = xr[d0 + 1], r2 = xr[d0 + 2], r3 = xr[d0 + 3];
    const float c0 = xc[d0 + 0], c1 = xc[d0 + 1], c2 = xc[d0 + 2], c3 = xc[d0 + 3];
    float dot = r0 * c0 + r1 * c1 + r2 * c2 + r3 * c3;
    float nr  = r0 * r0 + r1 * r1 + r2 * r2 + r3 * r3;
    float nc  = c0 * c0 + c1 * c1 + c2 * c2 + c3 * c3;
    if (lane == 0) dot += xr[128] * xc[128];  // homogeneous coord: dot only
#pragma unroll
    for (int off = 16; off >= 1; off >>= 1) {
        dot += __shfl_xor(dot, off, 32);
        nr  += __shfl_xor(nr,  off, 32);
        nc  += __shfl_xor(nc,  off, 32);
    }
    const float q = dot * dot / ((nr + EPSF) * (nc + EPSF));
    const float w = __expf(q - 1.0f);         // exp(-(1 - q))
    float* arow = acc + (long)row * DP1;
    atomicAdd(arow + d0 + 0, w * c0);
    atomicAdd(arow + d0 + 1, w * c1);
    atomicAdd(arow + d0 + 2, w * c2);
    atomicAdd(arow + d0 + 3, w * c3);
    if (lane == 0) atomicAdd(arow + 128, w);
}

// ---------------------------------------------------------------------------
// Kernel 3: fused scale + dual FP32 WMMA GEMM + ReLU + homogeneous column.
// Block = 8 waves, owns 16 nodes; wave w owns output columns [16w, 16w+16).
// A tiles (scaled neigh + self features) staged in LDS, stride 132 to spread
// the m-strided fragment reads across banks (bank = 4m + k).
// ---------------------------------------------------------------------------
__global__ __launch_bounds__(256) void uhg_fused_gemm(
    const float* __restrict__ x, const float* __restrict__ Wn,
    const float* __restrict__ Ws, float* __restrict__ out, int N) {
    __shared__ float As[2][16][132];
    __shared__ float scl[16];
    const int t = threadIdx.x;
    const int node0 = blockIdx.x * 16;

    if (t < 16) {
        int r = node0 + t; if (r >= N) r = N - 1;
        const float wsum = out[(long)r * DP1 + 128];
        scl[t] = 1.0f / fmaxf(wsum, WMINF);
    }
    __syncthreads();
#pragma unroll
    for (int i = t; i < 16 * DIM; i += 256) {
        const int r = i >> 7, k = i & 127;
        int rg = node0 + r; if (rg >= N) rg = N - 1;
        const long base = (long)rg * DP1 + k;
        As[0][r][k] = out[base] * scl[r];     // neigh_features (scaled)
        As[1][r][k] = x[base];                // self features
    }
    __syncthreads();

    const int wave = t >> 5;
    const int lane = t & 31;
    const int n0   = wave * 16;
    const int m    = lane & 15;               // A row / B column within tile
    const int koff = (lane >> 4) << 1;        // 0 or 2 (ISA f32 A/B layout)
    const float* wn = Wn + (long)(n0 + m) * DIM;   // B[k][n] = W[n][k]
    const float* ws = Ws + (long)(n0 + m) * DIM;
    __builtin_prefetch(wn, 0, 1);             // global_prefetch_b8: warm B rows
    __builtin_prefetch(ws, 0, 1);

    v8f c = {};
#pragma unroll 4
    for (int k = 0; k < DIM; k += 4) {
        v2f a, b;
        a[0] = As[0][m][k + koff]; a[1] = As[0][m][k + koff + 1];
        b[0] = wn[k + koff];       b[1] = wn[k + koff + 1];
        c = __builtin_amdgcn_wmma_f32_16x16x4_f32(false, a, false, b,
                                                  (short)0, c, false, false);
        a[0] = As[1][m][k + koff]; a[1] = As[1][m][k + koff + 1];
        b[0] = ws[k + koff];       b[1] = ws[k + koff + 1];
        c = __builtin_amdgcn_wmma_f32_16x16x4_f32(false, a, false, b,
                                                  (short)0, c, false, false);
    }

    // C/D layout: vgpr r -> row m = r + 8*(lane>>4), col n = lane&15
    const int colg  = n0 + m;
    const int rbase = node0 + ((lane >> 4) << 3);
    float* orow = out + (long)rbase * DP1 + colg;
    if (rbase + 7 < N) {
        // Fast path (always taken when N % 16 == 0): branch-free strip store.
#pragma unroll
        for (int r = 0; r < 8; ++r)
            orow[(long)r * DP1] = fmaxf(c[r], 0.0f);
    } else {
#pragma unroll
        for (int r = 0; r < 8; ++r)
            if (rbase + r < N) orow[(long)r * DP1] = fmaxf(c[r], 0.0f);
    }
    if (wave == 0 && lane < 16) {
        const int rg = node0 + lane;
        if (rg < N) {
            const long idx = (long)rg * DP1 + 128;
            out[idx] = fmaxf(x[idx], 0.0f);   // relu(homogeneous)
        }
    }
}

// ---------------------------------------------------------------------------
extern "C" void kernel_launch(void* const* d_in, const int* in_sizes, int n_in,
                              void* d_out, int out_size, void* d_ws, size_t ws_size,
                              hipStream_t stream) {
    (void)n_in; (void)out_size; (void)d_ws; (void)ws_size;
    const float* x  = (const float*)d_in[0];
    const int*   ei = (const int*)d_in[1];
    const float* Wn = (const float*)d_in[2];
    const float* Ws = (const float*)d_in[3];
    float* out = (float*)d_out;
    const int N = in_sizes[0] / DP1;
    const int E = in_sizes[1] / 2;

    const long total = (long)N * DP1;
    uhg_zero_f32<<<(int)((total + 255) / 256), 256, 0, stream>>>(out, total);
    uhg_edge_scatter<<<(E + 7) / 8, 256, 0, stream>>>(x, ei, out, E);
    uhg_fused_gemm<<<(N + 15) / 16, 256, 0, stream>>>(x, Wn, Ws, out, N);
}